// OCRGCNBase_85220741087383
// MI455X (gfx1250) — compile-verified
//
#include <hip/hip_runtime.h>

#define N_NODES 50000
#define N_EDGES 1600000
#define DIM     128
#define N_REL   8
#define N_LAYERS 3
#define KBIG    (DIM * (N_REL + 1))   /* 1152 */
#define KCHUNK  576                   /* 2 chunks of 576 = 1152 */
#define LDS_STRIDE 580                /* 580 % 64 == 4 -> conflict-free b64 A reads */

typedef __attribute__((ext_vector_type(2))) float v2f;
typedef __attribute__((ext_vector_type(8))) float v8f;

// ---------------------------------------------------------------------------
// Zero-fill (float4 granularity, grid-stride)
// ---------------------------------------------------------------------------
__global__ __launch_bounds__(256) void fill_zero_kernel(float4* __restrict__ p, long n4) {
    long i = (long)blockIdx.x * blockDim.x + threadIdx.x;
    long stride = (long)gridDim.x * blockDim.x;
    float4 z = make_float4(0.f, 0.f, 0.f, 0.f);
    for (; i < n4; i += stride) p[i] = z;
}

// ---------------------------------------------------------------------------
// Edge scatter: one wave32 per edge.
// agg[dst][rel][0:128] += h[src][0:128]; cnt[dst][rel] += 1
// ---------------------------------------------------------------------------
__global__ __launch_bounds__(256) void scatter_kernel(
    const float* __restrict__ h,
    const int*   __restrict__ src,
    const int*   __restrict__ dst,
    const int*   __restrict__ et,
    float*       __restrict__ agg,
    float*       __restrict__ cnt) {
    int e    = blockIdx.x * 8 + (threadIdx.x >> 5);
    int lane = threadIdx.x & 31;
    if (e >= N_EDGES) return;
    int s = src[e];
    int d = dst[e];
    int r = et[e];
    const float4 v = *(const float4*)(h + (size_t)s * DIM + lane * 4);
    float* base = agg + (((size_t)d * N_REL) + r) * DIM + lane * 4;
    unsafeAtomicAdd(base + 0, v.x);
    unsafeAtomicAdd(base + 1, v.y);
    unsafeAtomicAdd(base + 2, v.z);
    unsafeAtomicAdd(base + 3, v.w);
    if (lane == 0) unsafeAtomicAdd(cnt + (size_t)d * N_REL + r, 1.0f);
}

// ---------------------------------------------------------------------------
// cnt -> 1 / max(cnt, 1)  (in place)
// ---------------------------------------------------------------------------
__global__ __launch_bounds__(256) void invcnt_kernel(float* __restrict__ cnt) {
    int i = blockIdx.x * blockDim.x + threadIdx.x;
    if (i < N_NODES * N_REL) {
        float c = cnt[i];
        cnt[i] = 1.0f / fmaxf(c, 1.0f);
    }
}

// ---------------------------------------------------------------------------
// Cooperative A-panel stage: 16 rows x KCHUNK cols of Xbig into LDS.
// Xbig[i][k]: k<128 -> h[i][k]; else -> agg[i][r][kr] * invc[i][r]
// ---------------------------------------------------------------------------
__device__ __forceinline__ void stage_panel(
    float* __restrict__ lA, int tid, int m0, int kbaseg,
    const float* __restrict__ h_in,
    const float* __restrict__ agg,
    const float* __restrict__ invc) {
    for (int idx = tid; idx < 16 * (KCHUNK / 4); idx += 256) {
        int m   = idx / (KCHUNK / 4);
        int kl  = (idx % (KCHUNK / 4)) * 4;   // local column
        int kg  = kbaseg + kl;                // global column in Xbig
        int row = m0 + m;
        float4 v = make_float4(0.f, 0.f, 0.f, 0.f);
        if (row < N_NODES) {
            if (kg < DIM) {
                v = *(const float4*)(h_in + (size_t)row * DIM + kg);
            } else {
                int kk = kg - DIM;
                int r  = kk >> 7;
                int kr = kk & 127;
                float s  = invc[(size_t)row * N_REL + r];
                float4 a = *(const float4*)(agg + ((size_t)row * N_REL + r) * DIM + kr);
                v.x = a.x * s; v.y = a.y * s; v.z = a.z * s; v.w = a.w * s;
            }
        }
        *(float4*)(&lA[m * LDS_STRIDE + kl]) = v;
    }
}

// ---------------------------------------------------------------------------
// Fused RGCN layer GEMM via fp32 WMMA:
//   out[i][j] = bias[j] + sum_k Xbig[i][k] * Wbig[k][j]   (ReLU optional)
// Wbig = [ W_root (128x128) ; W_rel (1024x128) ]  (row-major)
// Block = 256 threads = 8 waves; block owns 16 rows; wave w owns cols [16w,16w+16).
// K-loops are split at the root/rel boundary so the weight base pointer is
// loop-invariant -> B fetches become base + immediate offsets (no branches).
// ---------------------------------------------------------------------------
__global__ __launch_bounds__(256) void rgcn_gemm_kernel(
    const float* __restrict__ h_in,    // [N,128]
    const float* __restrict__ agg,     // [N,R,128]
    const float* __restrict__ invc,    // [N,R]
    const float* __restrict__ w_root,  // [128,128]
    const float* __restrict__ w_rel,   // [1024,128]
    const float* __restrict__ bias,    // [128]
    float*       __restrict__ out,     // [N,128]
    int relu) {
    __shared__ float lA[16 * LDS_STRIDE];

    const int tid  = threadIdx.x;
    const int m0   = blockIdx.x * 16;
    const int wave = tid >> 5;
    const int lane = tid & 31;
    const int half = lane >> 4;
    const int l16  = lane & 15;
    const int j    = wave * 16 + l16;          // output column

    v8f acc0 = {0.f, 0.f, 0.f, 0.f, 0.f, 0.f, 0.f, 0.f};
    v8f acc1 = {0.f, 0.f, 0.f, 0.f, 0.f, 0.f, 0.f, 0.f};
    const float* lArow = &lA[l16 * LDS_STRIDE];
    const int h2 = 2 * half;

    // ---------------- chunk 0: Xbig cols [0, 576) ----------------
    stage_panel(lA, tid, m0, 0, h_in, agg, invc);
    __syncthreads();

    {   // root part: k in [0, 128)
        const float* Wp = w_root + (size_t)h2 * DIM + j;
        #pragma unroll
        for (int k0 = 0; k0 < DIM; k0 += 8) {
            v2f a0 = *(const v2f*)(lArow + k0 + h2);
            v2f b0;
            b0.x = Wp[(size_t)(k0 + 0) * DIM];
            b0.y = Wp[(size_t)(k0 + 1) * DIM];
            acc0 = __builtin_amdgcn_wmma_f32_16x16x4_f32(
                false, a0, false, b0, (short)0, acc0, false, false);
            v2f a1 = *(const v2f*)(lArow + k0 + 4 + h2);
            v2f b1;
            b1.x = Wp[(size_t)(k0 + 4) * DIM];
            b1.y = Wp[(size_t)(k0 + 5) * DIM];
            acc1 = __builtin_amdgcn_wmma_f32_16x16x4_f32(
                false, a1, false, b1, (short)0, acc1, false, false);
        }
    }
    {   // relation rows [0, 448): k in [128, 576)
        const float* Wp = w_rel + (size_t)h2 * DIM + j;
        #pragma unroll 4
        for (int k0 = DIM; k0 < KCHUNK; k0 += 8) {
            int kk = k0 - DIM;
            v2f a0 = *(const v2f*)(lArow + k0 + h2);
            v2f b0;
            b0.x = Wp[(size_t)(kk + 0) * DIM];
            b0.y = Wp[(size_t)(kk + 1) * DIM];
            acc0 = __builtin_amdgcn_wmma_f32_16x16x4_f32(
                false, a0, false, b0, (short)0, acc0, false, false);
            v2f a1 = *(const v2f*)(lArow + k0 + 4 + h2);
            v2f b1;
            b1.x = Wp[(size_t)(kk + 4) * DIM];
            b1.y = Wp[(size_t)(kk + 5) * DIM];
            acc1 = __builtin_amdgcn_wmma_f32_16x16x4_f32(
                false, a1, false, b1, (short)0, acc1, false, false);
        }
    }
    __syncthreads();

    // ---------------- chunk 1: Xbig cols [576, 1152) ----------------
    stage_panel(lA, tid, m0, KCHUNK, h_in, agg, invc);
    __syncthreads();

    {   // relation rows [448, 1024): k in [576, 1152)
        const float* Wp = w_rel + (size_t)(KCHUNK - DIM + h2) * DIM + j;
        #pragma unroll 4
        for (int k0 = 0; k0 < KCHUNK; k0 += 8) {
            v2f a0 = *(const v2f*)(lArow + k0 + h2);
            v2f b0;
            b0.x = Wp[(size_t)(k0 + 0) * DIM];
            b0.y = Wp[(size_t)(k0 + 1) * DIM];
            acc0 = __builtin_amdgcn_wmma_f32_16x16x4_f32(
                false, a0, false, b0, (short)0, acc0, false, false);
            v2f a1 = *(const v2f*)(lArow + k0 + 4 + h2);
            v2f b1;
            b1.x = Wp[(size_t)(k0 + 4) * DIM];
            b1.y = Wp[(size_t)(k0 + 5) * DIM];
            acc1 = __builtin_amdgcn_wmma_f32_16x16x4_f32(
                false, a1, false, b1, (short)0, acc1, false, false);
        }
    }

    v8f acc = acc0 + acc1;

    // Epilogue. C layout: lanes 0-15 vgpr v -> M=v ; lanes 16-31 vgpr v -> M=8+v.
    float bj = bias[j];
    #pragma unroll
    for (int v = 0; v < 8; ++v) {
        int row = m0 + v + half * 8;
        if (row < N_NODES) {
            float o = acc[v] + bj;
            if (relu) o = fmaxf(o, 0.f);
            out[(size_t)row * DIM + j] = o;
        }
    }
}

// ---------------------------------------------------------------------------
// Host launcher
// ---------------------------------------------------------------------------
extern "C" void kernel_launch(void* const* d_in, const int* in_sizes, int n_in,
                              void* d_out, int out_size, void* d_ws, size_t ws_size,
                              hipStream_t stream) {
    const float* x   = (const float*)d_in[0];
    const int*   ei  = (const int*)  d_in[1];
    const int*   et  = (const int*)  d_in[2];
    const float* Ws  = (const float*)d_in[3];   // [L,R,D,D]
    const float* Wr  = (const float*)d_in[4];   // [L,D,D]
    const float* bs  = (const float*)d_in[5];   // [L,D]
    float* out = (float*)d_out;

    float* ws  = (float*)d_ws;
    float* agg = ws;                                          // N*R*D
    float* cnt = agg + (size_t)N_NODES * N_REL * DIM;         // N*R (becomes inv-count)
    float* hA  = cnt + (size_t)N_NODES * N_REL;               // N*D
    float* hB  = hA  + (size_t)N_NODES * DIM;                 // N*D

    const int* src = ei;
    const int* dst = ei + N_EDGES;

    const long nz4 = ((long)N_NODES * N_REL * DIM + (long)N_NODES * N_REL) / 4;

    const float* cur = x;
    for (int layer = 0; layer < N_LAYERS; ++layer) {
        float* nxt = (layer == 0) ? hA : (layer == 1 ? hB : out);

        fill_zero_kernel<<<4096, 256, 0, stream>>>((float4*)agg, nz4);

        scatter_kernel<<<(N_EDGES + 7) / 8, 256, 0, stream>>>(
            cur, src, dst, et, agg, cnt);

        invcnt_kernel<<<(N_NODES * N_REL + 255) / 256, 256, 0, stream>>>(cnt);

        rgcn_gemm_kernel<<<(N_NODES + 15) / 16, 256, 0, stream>>>(
            cur, agg, cnt,
            Wr + (size_t)layer * DIM * DIM,
            Ws + (size_t)layer * N_REL * DIM * DIM,
            bs + (size_t)layer * DIM,
            nxt, (layer < N_LAYERS - 1) ? 1 : 0);

        cur = nxt;
    }
}